// AssignmentLoss_83511344103781
// MI455X (gfx1250) — compile-verified
//
#include <hip/hip_runtime.h>
#include <hip/hip_bf16.h>

typedef __attribute__((ext_vector_type(2))) float v2f;
typedef __attribute__((ext_vector_type(8))) float v8f;

#define DIM   512
#define KSTEP (DIM / 4)   // 128 WMMA steps of K=4
#define EPSN  1e-8f

// One wave handles 16 samples. A (16x4) and B (4x16) f32 WMMA operands have the
// SAME per-lane layout: lane l supplies row m=l&15, cols 4t + 2*(l>>4) + {0,1}.
// So a single float2 load per matrix per K-step feeds the matrix pipe directly.
// diag(C) after 128 steps = e_i . p_i ; VALU accumulates |e|^2, |p|^2 in parallel.
__global__ __launch_bounds__(256) void cosloss_main_kernel(
    const float* __restrict__ emb,
    const long long* __restrict__ labels,
    const float* __restrict__ protos,
    float* __restrict__ partials)
{
    const int tid  = threadIdx.x;
    const int wave = tid >> 5;          // 0..7
    const int lane = tid & 31;
    const int m    = lane & 15;         // sample index within 16-group
    const int o    = (lane >> 4) * 2;   // K sub-offset: 0 (lanes 0-15) or 2 (16-31)

    const int group  = blockIdx.x * 8 + wave;
    const int sample = group * 16 + m;

    const long long lab = labels[sample];
    const float* erow = emb    + (size_t)sample * DIM;
    const float* prow = protos + (size_t)lab    * DIM;

    v8f acc = {0.f, 0.f, 0.f, 0.f, 0.f, 0.f, 0.f, 0.f};
    float see = 0.f, spp = 0.f;

#pragma unroll 4
    for (int t = 0; t < KSTEP; ++t) {
        const int col = 4 * t + o;
        v2f a = *(const v2f*)(erow + col);
        v2f b = *(const v2f*)(prow + col);
        // D = A(16x4) * B(4x16) + C  -> v_wmma_f32_16x16x4_f32
        acc = __builtin_amdgcn_wmma_f32_16x16x4_f32(
            /*neg_a=*/false, a, /*neg_b=*/false, b,
            /*c_mod=*/(short)0, acc, /*reuse_a=*/false, /*reuse_b=*/false);
        // norms ride on the VALU pipe, co-executing with the XDL WMMA
        see = fmaf(a.x, a.x, fmaf(a.y, a.y, see));
        spp = fmaf(b.x, b.x, fmaf(b.y, b.y, spp));
    }

    // Lanes l and l^16 each covered half of row m's columns: combine.
    see += __shfl_xor(see, 16, 32);
    spp += __shfl_xor(spp, 16, 32);

    // Diagonal extraction via masked accumulation (pure v_cmp/v_cndmask/v_add,
    // no control flow — EXEC stays all-ones):
    //   sample r   (r<8): C[r][r]     at VGPR r, lane r       (lanes 0-7)
    //   sample r+8      : C[r+8][r+8] at VGPR r, lane r+24    (lanes 24-31)
    // Lanes 8..23 match no r -> dv = 0 -> contribute 0 to the sum.
    float dv = 0.f;
#pragma unroll
    for (int r = 0; r < 8; ++r)
        dv += ((lane == r) || (lane == r + 24)) ? acc[r] : 0.f;

    float en = fmaxf(sqrtf(see), EPSN);   // torch cosine_similarity eps clamp
    float pn = fmaxf(sqrtf(spp), EPSN);
    float c  = dv / (en * pn);            // safe: en*pn >= eps^2 > 0

    // Sum the 16 cosine values across the wave (fixed xor-tree: deterministic).
#pragma unroll
    for (int mask = 16; mask > 0; mask >>= 1)
        c += __shfl_xor(c, mask, 32);

    __shared__ float wsum[8];
    if (lane == 0) wsum[wave] = c;
    __syncthreads();
    if (tid == 0) {
        float s = 0.f;
#pragma unroll
        for (int i = 0; i < 8; ++i) s += wsum[i];
        partials[blockIdx.x] = s;
    }
}

// Single-block, fixed-order tree reduction -> deterministic scalar.
__global__ __launch_bounds__(256) void cosloss_reduce_kernel(
    const float* __restrict__ partials, int n, float* __restrict__ out, float invB)
{
    __shared__ float sm[256];
    const int tid = threadIdx.x;
    float v = 0.f;
    for (int i = tid; i < n; i += 256) v += partials[i];
    sm[tid] = v;
    __syncthreads();
#pragma unroll
    for (int s = 128; s > 0; s >>= 1) {
        if (tid < s) sm[tid] += sm[tid + s];
        __syncthreads();
    }
    if (tid == 0) out[0] = 1.0f - sm[0] * invB;
}

extern "C" void kernel_launch(void* const* d_in, const int* in_sizes, int n_in,
                              void* d_out, int out_size, void* d_ws, size_t ws_size,
                              hipStream_t stream) {
    const float*     emb    = (const float*)d_in[0];      // [B, 512] f32
    const long long* labels = (const long long*)d_in[1];  // [B] i64
    const float*     protos = (const float*)d_in[2];      // [C, 512] f32
    float*           out    = (float*)d_out;              // scalar f32
    float*           parts  = (float*)d_ws;

    const int B      = in_sizes[1];        // 65536
    const int blocks = B / 128;            // 16 samples/wave * 8 waves/block

    cosloss_main_kernel<<<blocks, 256, 0, stream>>>(emb, labels, protos, parts);
    cosloss_reduce_kernel<<<1, 256, 0, stream>>>(parts, blocks, out, 1.0f / (float)B);
}